// LSSLfixed_42743514529984
// MI455X (gfx1250) — compile-verified
//
#include <hip/hip_runtime.h>

typedef __attribute__((ext_vector_type(2))) float v2f;
typedef __attribute__((ext_vector_type(8))) float v8f;
typedef __attribute__((ext_vector_type(4))) int   v4i;

typedef __attribute__((address_space(1))) v4i* gv4i_p;  // global 16B vector
typedef __attribute__((address_space(3))) v4i* lv4i_p;  // LDS    16B vector

#define N_STATE 128
#define H_CH    128
#define L_SEQ   2048

// ---------------------------------------------------------------------------
// Stage 1: Kl[h, l] = dot(C[h, :], K[l, h, :])
// One wave32 per (l,h) pair; lane loads float4 -> 512B fully-coalesced per
// wave from K_mats (the 128 MiB stream that dominates the roofline).
// ---------------------------------------------------------------------------
__global__ __launch_bounds__(256) void kl_synth_kernel(
    const float* __restrict__ C, const float* __restrict__ K,
    float* __restrict__ Kl) {
  const int lane = threadIdx.x & 31;
  const int wave = threadIdx.x >> 5;
  const int p = blockIdx.x * 8 + wave;      // p = l*H + h  (K_mats row index)
  const int h = p & (H_CH - 1);
  const int l = p >> 7;                     // p / H

  const float4 kv = *(const float4*)(K + (size_t)p * N_STATE + lane * 4);
  const float4 cv = *(const float4*)(C + h * N_STATE + lane * 4);
  float part = kv.x * cv.x + kv.y * cv.y + kv.z * cv.z + kv.w * cv.w;

#pragma unroll
  for (int off = 16; off >= 1; off >>= 1)
    part += __shfl_xor(part, off, 32);

  if (lane == 0) Kl[h * L_SEQ + l] = part;  // store channel-major for stage 2
}

// ---------------------------------------------------------------------------
// 16-byte global -> LDS copy via GLOBAL_LOAD_ASYNC_TO_LDS_B128 (ASYNCcnt).
// Signature (probe-confirmed by round-2 diagnostic): arg0 = v4i addrspace(1)*,
// arg1 = v4i addrspace(3)*, then imm offset and imm cpol.
// ---------------------------------------------------------------------------
__device__ __forceinline__ void copy16_g2l(float* lds_dst, const float* gsrc) {
#if __has_builtin(__builtin_amdgcn_global_load_async_to_lds_b128)
  __builtin_amdgcn_global_load_async_to_lds_b128(
      (gv4i_p)(gsrc), (lv4i_p)(lds_dst), /*imm offset=*/0, /*cpol=*/0);
#else
  *(float4*)lds_dst = *(const float4*)gsrc;
#endif
}

__device__ __forceinline__ void wait_async_then_barrier() {
#if __has_builtin(__builtin_amdgcn_global_load_async_to_lds_b128)
#if __has_builtin(__builtin_amdgcn_s_wait_asynccnt)
  __builtin_amdgcn_s_wait_asynccnt(0);
#else
  asm volatile("s_wait_asynccnt 0" ::: "memory");
#endif
#endif
  __syncthreads();
}

// ---------------------------------------------------------------------------
// Stage 2: depthwise 'same' conv + skip, as dense V_WMMA_F32_16X16X4_F32.
//   D[a][n] = y[h, i0 + a + 16n]
//   A[a][b] = Kl[h, bm + a - b]                (Toeplitz band, 16x4)
//   B[b][n] = u_pad[h, i0 + 1023 - bm + b + 16n]
// bm sweeps 0..2048 step 4: every MAC is a useful conv tap (zero waste).
// One block per channel; 8 waves = 8 output tiles of 256. LDS guard bands
// implement the zero padding of both Kl and u.
// ---------------------------------------------------------------------------
#define KOFF 4
#define KPAD 16
#define KLDS (KOFF + L_SEQ + KPAD)   // indices [-4, 2063]  -> 2068 floats
#define UOFF 1056
#define UPAD 1280
#define ULDS (UOFF + L_SEQ + UPAD)   // indices [-1056, 3327] -> 4384 floats

__global__ __launch_bounds__(256) void conv_wmma_kernel(
    const float* __restrict__ u, const float* __restrict__ D,
    const float* __restrict__ Kl, float* __restrict__ y) {
  __shared__ __align__(16) float lds_k[KLDS];
  __shared__ __align__(16) float lds_u[ULDS];

  const int h   = blockIdx.x;
  const int tid = threadIdx.x;

  // Zero only the guard bands (disjoint from the async-written payload).
  if (tid < KOFF) lds_k[tid] = 0.0f;                         // Kl[-4..-1]
  if (tid < KPAD) lds_k[KOFF + L_SEQ + tid] = 0.0f;          // Kl[2048..2063]
  for (int i = tid; i < UOFF; i += 256) lds_u[i] = 0.0f;     // u[-1056..-1]
  for (int i = tid; i < UPAD; i += 256) lds_u[UOFF + L_SEQ + i] = 0.0f;

  // Async-stage the payloads: 16B per lane, global -> LDS (ASYNCcnt path).
  for (int i = tid; i < L_SEQ / 4; i += 256) {
    copy16_g2l(&lds_k[KOFF + 4 * i], Kl + h * L_SEQ + 4 * i);
    copy16_g2l(&lds_u[UOFF + 4 * i], u + h * L_SEQ + 4 * i);
  }
  wait_async_then_barrier();

  const int wave = tid >> 5;
  const int lane = tid & 31;
  const int l15  = lane & 15;          // N column (B/D) and M row (A)
  const int half = lane >> 4;
  const int k0   = half << 1;          // K rows {0,1} or {2,3} per ISA layout
  const int i0   = wave * 256;         // this wave's output tile base

  v8f c = {0.f, 0.f, 0.f, 0.f, 0.f, 0.f, 0.f, 0.f};

  const float* kp = lds_k + (KOFF + l15 - k0);
  const float* up = lds_u + (UOFF + i0 + 1023 + k0 + 16 * l15);

#pragma unroll 4
  for (int bm = 0; bm <= 2048; bm += 4) {
    v2f a, b;
    a.x = kp[bm];          // A[M=l15][K=k0]   = Kl[bm + l15 - k0]
    a.y = kp[bm - 1];      // A[M=l15][K=k0+1] = Kl[bm + l15 - k0 - 1]
    b.x = up[-bm];         // B[K=k0][N=l15]   = u_pad[i0+1023-bm + k0   + 16*l15]
    b.y = up[-bm + 1];     // B[K=k0+1][N=l15] = u_pad[i0+1023-bm + k0+1 + 16*l15]
    c = __builtin_amdgcn_wmma_f32_16x16x4_f32(
        /*neg_a=*/false, a, /*neg_b=*/false, b,
        /*c_mod=*/(short)0, c, /*reuse_a=*/false, /*reuse_b=*/false);
  }

  const float dh = D[h];
#pragma unroll
  for (int v = 0; v < 8; ++v) {
    const int idx = i0 + (v + 8 * half) + 16 * l15;   // D-matrix lane layout
    y[h * L_SEQ + idx] = c[v] + dh * lds_u[UOFF + idx];
  }
}

// ---------------------------------------------------------------------------
extern "C" void kernel_launch(void* const* d_in, const int* in_sizes, int n_in,
                              void* d_out, int out_size, void* d_ws,
                              size_t ws_size, hipStream_t stream) {
  const float* u = (const float*)d_in[0];  // (H, L)
  const float* C = (const float*)d_in[1];  // (H, N)
  const float* D = (const float*)d_in[2];  // (H,)
  const float* K = (const float*)d_in[3];  // (L, H, N)
  float* y  = (float*)d_out;               // (H, L)
  float* Kl = (float*)d_ws;                // (H, L) scratch, 1 MiB

  kl_synth_kernel<<<(L_SEQ * H_CH) / 8, 256, 0, stream>>>(C, K, Kl);
  conv_wmma_kernel<<<H_CH, 256, 0, stream>>>(u, D, Kl, y);
}